// GraphSAGEEncoder_34634616274989
// MI455X (gfx1250) — compile-verified
//
#include <hip/hip_runtime.h>
#include <hip/hip_bf16.h>

typedef __attribute__((ext_vector_type(2))) float v2f;
typedef __attribute__((ext_vector_type(8))) float v8f;

// ---------------------------------------------------------------------------
// Kernel 0: zero the aggregation buffer [N*32] and degree buffer [N]
// ---------------------------------------------------------------------------
__global__ __launch_bounds__(256) void sage_init(float* __restrict__ agg,
                                                 float* __restrict__ deg,
                                                 int n_nodes) {
  const long long t = (long long)blockIdx.x * blockDim.x + threadIdx.x;
  const long long n_agg = (long long)n_nodes * 32;
  if (t < n_agg) {
    agg[t] = 0.0f;
  } else if (t < n_agg + n_nodes) {
    deg[t - n_agg] = 0.0f;
  }
}

// ---------------------------------------------------------------------------
// Kernel 1: dual-weight projection GEMM via V_WMMA_F32_16X16X4_F32.
//   Yl = X @ Wl   and   Yr = X @ Wr,   X:[N,K] row-major, W:[K,32] row-major.
// One wave32 per 16-row M tile; 4 f32 accumulators (2 N-tiles x 2 weights).
// A layout (16x4 f32): lanes 0-15 -> row=lane, {v0,v1}={K=k,K=k+1};
//                      lanes16-31 -> row=lane-16, {v0,v1}={K=k+2,K=k+3}.
// B layout (4x16 f32): mirrored K striping across lane halves, N = lane&15.
// C/D layout (16x16 f32): vgpr j -> M = j + 8*(lane>=16), N = lane&15.
// ---------------------------------------------------------------------------
__global__ __launch_bounds__(256) void sage_gemm(
    const float* __restrict__ X, int K, int n_nodes,
    const float* __restrict__ Wl, const float* __restrict__ Wr,
    float* __restrict__ Yl, float* __restrict__ Yr) {
  const int wavesPerBlock = blockDim.x >> 5;
  const int tile = blockIdx.x * wavesPerBlock + (threadIdx.x >> 5);
  const int numTiles = (n_nodes + 15) >> 4;
  if (tile >= numTiles) return;  // wave-uniform: EXEC stays all-1s below

  const int lane = threadIdx.x & 31;
  const int hi   = lane >> 4;      // 0: lanes 0-15, 1: lanes 16-31
  const int l15  = lane & 15;
  const int koff = hi * 2;

  int row = tile * 16 + l15;
  if (row >= n_nodes) row = n_nodes - 1;  // clamp (N=100000 is 16-aligned)
  const float* xrow = X + (long long)row * K;

  v8f acc_l0 = {}, acc_l1 = {}, acc_r0 = {}, acc_r1 = {};

  for (int k = 0; k < K; k += 4) {
    v2f a;
    a.x = xrow[k + koff];
    a.y = xrow[k + koff + 1];

    const int r0 = (k + koff) * 32;
    const int r1 = (k + koff + 1) * 32;
    v2f bl0, bl1, br0, br1;
    bl0.x = Wl[r0 + l15];       bl0.y = Wl[r1 + l15];
    bl1.x = Wl[r0 + 16 + l15];  bl1.y = Wl[r1 + 16 + l15];
    br0.x = Wr[r0 + l15];       br0.y = Wr[r1 + l15];
    br1.x = Wr[r0 + 16 + l15];  br1.y = Wr[r1 + 16 + l15];

    acc_l0 = __builtin_amdgcn_wmma_f32_16x16x4_f32(false, a, false, bl0,
                                                   (short)0, acc_l0, false, false);
    acc_l1 = __builtin_amdgcn_wmma_f32_16x16x4_f32(false, a, false, bl1,
                                                   (short)0, acc_l1, false, false);
    acc_r0 = __builtin_amdgcn_wmma_f32_16x16x4_f32(false, a, false, br0,
                                                   (short)0, acc_r0, false, false);
    acc_r1 = __builtin_amdgcn_wmma_f32_16x16x4_f32(false, a, false, br1,
                                                   (short)0, acc_r1, false, false);
  }

  const int m0 = tile * 16 + hi * 8;
#pragma unroll
  for (int j = 0; j < 8; ++j) {
    const int m = m0 + j;
    if (m < n_nodes) {
      const long long o = (long long)m * 32;
      Yl[o + l15]      = acc_l0[j];
      Yl[o + 16 + l15] = acc_l1[j];
      Yr[o + l15]      = acc_r0[j];
      Yr[o + 16 + l15] = acc_r1[j];
    }
  }
}

// ---------------------------------------------------------------------------
// Kernel 2: edge scatter. One wave32 per edge: lane c handles channel c.
//   agg[dst][c] += Y[src][c]   (L2-resident fp32 atomics)
// Optionally accumulates degree (layer 1 only).
// ---------------------------------------------------------------------------
__global__ __launch_bounds__(256) void sage_scatter(
    const long long* __restrict__ src, const long long* __restrict__ dst,
    const float* __restrict__ Y, float* __restrict__ agg,
    float* __restrict__ deg, int add_deg, int n_edges) {
  const long long t = (long long)blockIdx.x * blockDim.x + threadIdx.x;
  const int e = (int)(t >> 5);
  if (e >= n_edges) return;
  const int c = (int)(t & 31);
  const int s = (int)src[e];
  const int d = (int)dst[e];
  const float v = Y[(long long)s * 32 + c];
  atomicAdd(&agg[(long long)d * 32 + c], v);
  if (add_deg && c == 0) atomicAdd(&deg[d], 1.0f);
}

// ---------------------------------------------------------------------------
// Kernel 3: finish a SAGE layer:
//   out[n][c] = relu(agg[n][c] / max(deg[n],1) + bias[c] + Yr[n][c])
// Optionally re-zeros agg for the next layer (fused to save a pass).
// ---------------------------------------------------------------------------
__global__ __launch_bounds__(256) void sage_finish(
    const float* __restrict__ agg, const float* __restrict__ deg,
    const float* __restrict__ bias, const float* __restrict__ Yr,
    float* __restrict__ out, float* __restrict__ agg_zero, int n_nodes) {
  const long long t = (long long)blockIdx.x * blockDim.x + threadIdx.x;
  if (t >= (long long)n_nodes * 32) return;
  const int n = (int)(t >> 5);
  const int c = (int)(t & 31);
  const float inv = 1.0f / fmaxf(deg[n], 1.0f);
  const float v = agg[t] * inv + bias[c] + Yr[t];
  out[t] = fmaxf(v, 0.0f);
  if (agg_zero) agg_zero[t] = 0.0f;
}

// ---------------------------------------------------------------------------
// Host-side launch sequence (graph-capture safe, deterministic).
// ---------------------------------------------------------------------------
extern "C" void kernel_launch(void* const* d_in, const int* in_sizes, int n_in,
                              void* d_out, int out_size, void* d_ws, size_t ws_size,
                              hipStream_t stream) {
  const float*     x    = (const float*)d_in[0];          // [N,128]
  const long long* ei   = (const long long*)d_in[1];      // [2,E] int64
  const float*     W1l  = (const float*)d_in[2];          // [128,32]
  const float*     b1   = (const float*)d_in[3];          // [32]
  const float*     W1r  = (const float*)d_in[4];          // [128,32]
  const float*     W2l  = (const float*)d_in[5];          // [32,32]
  const float*     b2   = (const float*)d_in[6];          // [32]
  const float*     W2r  = (const float*)d_in[7];          // [32,32]
  float*           out  = (float*)d_out;                  // [N,32]

  const int N = in_sizes[0] / 128;   // 100000
  const int E = in_sizes[1] / 2;     // 1600000
  const long long* src = ei;
  const long long* dst = ei + E;

  // Workspace layout (floats): yl[N*32] | yr[N*32] | agg[N*32] | h[N*32] | deg[N]
  float* ws  = (float*)d_ws;
  float* yl  = ws;
  float* yr  = yl  + (long long)N * 32;
  float* agg = yr  + (long long)N * 32;
  float* h   = agg + (long long)N * 32;
  float* deg = h   + (long long)N * 32;

  const int BLK = 256;
  const int numTiles   = (N + 15) / 16;
  const int gemmBlocks = (numTiles + (BLK / 32) - 1) / (BLK / 32);
  const long long scatterThreads = (long long)E * 32;
  const int scatterBlocks = (int)((scatterThreads + BLK - 1) / BLK);
  const long long nodeThreads = (long long)N * 32;
  const int nodeBlocks = (int)((nodeThreads + BLK - 1) / BLK);
  const int initBlocks = (int)((nodeThreads + N + BLK - 1) / BLK);

  // 0) zero agg + deg
  sage_init<<<initBlocks, BLK, 0, stream>>>(agg, deg, N);

  // Layer 1: project (K=128), aggregate, finish (re-zeroing agg)
  sage_gemm<<<gemmBlocks, BLK, 0, stream>>>(x, 128, N, W1l, W1r, yl, yr);
  sage_scatter<<<scatterBlocks, BLK, 0, stream>>>(src, dst, yl, agg, deg, 1, E);
  sage_finish<<<nodeBlocks, BLK, 0, stream>>>(agg, deg, b1, yr, h, agg, N);

  // Layer 2: project (K=32), aggregate, finish -> output
  sage_gemm<<<gemmBlocks, BLK, 0, stream>>>(h, 32, N, W2l, W2r, yl, yr);
  sage_scatter<<<scatterBlocks, BLK, 0, stream>>>(src, dst, yl, agg, deg, 0, E);
  sage_finish<<<nodeBlocks, BLK, 0, stream>>>(agg, deg, b2, yr, out, nullptr, N);
}